// Res_GCN_12764642804231
// MI455X (gfx1250) — compile-verified
//
#include <hip/hip_runtime.h>

typedef __attribute__((ext_vector_type(2))) float v2f;
typedef __attribute__((ext_vector_type(8))) float v8f;

#define N_NODES   100000
#define N_EDGES   1600000
#define D         128
#define FEAT_ELEMS (N_NODES * D)     // 12,800,000 floats
#define M_TILES   (N_NODES / 16)     // 6250 exactly

// ---------------------------------------------------------------------------
// zero: grid-stride fp32 clear (d_ws is poisoned 0xAA; must re-zero each call)
// ---------------------------------------------------------------------------
__global__ __launch_bounds__(256) void zero_kernel(float* __restrict__ p, int n) {
  int i = blockIdx.x * blockDim.x + threadIdx.x;
  int stride = gridDim.x * blockDim.x;
  for (; i < n; i += stride) p[i] = 0.0f;
}

// ---------------------------------------------------------------------------
// degree: one thread per edge, atomic count on destination node
// ---------------------------------------------------------------------------
__global__ __launch_bounds__(256) void degree_kernel(const int* __restrict__ dst,
                                                     float* __restrict__ deg) {
  int e = blockIdx.x * blockDim.x + threadIdx.x;
  if (e < N_EDGES) atomicAdd(&deg[dst[e]], 1.0f);
}

// ---------------------------------------------------------------------------
// scatter-add: one 32-lane wave per edge; lane l handles floats [4l, 4l+4).
// Gather of feat[src] row (512B) is fully coalesced across the wave; the
// destination row is hot in L2 (sum buffer = 51.2MB < 192MB L2), so the
// 4 float atomics per lane resolve at L2 atomic units.
// ---------------------------------------------------------------------------
__global__ __launch_bounds__(256) void scatter_kernel(const float* __restrict__ feat,
                                                      const int* __restrict__ src,
                                                      const int* __restrict__ dst,
                                                      float* __restrict__ sum) {
  long gid = (long)blockIdx.x * blockDim.x + threadIdx.x;
  int e    = (int)(gid >> 5);
  int lane = (int)(gid & 31);
  if (e >= N_EDGES) return;
  int s = src[e];
  int d = dst[e];
  float4 v = ((const float4*)(feat + (size_t)s * D))[lane];
  float* o = sum + (size_t)d * D + lane * 4;
  atomicAdd(o + 0, v.x);
  atomicAdd(o + 1, v.y);
  atomicAdd(o + 2, v.z);
  atomicAdd(o + 3, v.w);
}

// ---------------------------------------------------------------------------
// Fused SAGE layer: out = relu( (sum * 1/max(deg,1)) @ Wl + bl + xin @ Wr )
// Block = 128 threads = 4 waves. Each wave owns one 16x16 output tile:
//   M-tile = blockIdx.x*4 + wave, N-tile = blockIdx.y.
// Both 128x16 weight tiles staged in LDS (16KB). K stepped by 4 with
// v_wmma_f32_16x16x4_f32, two WMMAs per step (aggregate path + self path)
// chained through one f32 accumulator initialized with the bias.
// ---------------------------------------------------------------------------
__global__ __launch_bounds__(128) void sage_gemm_kernel(
    const float* __restrict__ sum,   // [N,128] unnormalized aggregate
    const float* __restrict__ deg,   // [N]
    const float* __restrict__ xin,   // [N,128] self features
    const float* __restrict__ Wl,    // [128,128] row-major (k,n)
    const float* __restrict__ bl,    // [128]
    const float* __restrict__ Wr,    // [128,128] row-major (k,n)
    float* __restrict__ out)         // [N,128]
{
  __shared__ float ldsWl[D * 16];
  __shared__ float ldsWr[D * 16];
  const int n_base = blockIdx.y * 16;

  // Cooperative weight-tile staging: ldsW[k*16 + n] = W[k][n_base + n]
  for (int i = threadIdx.x; i < D * 16; i += 128) {
    int kk = i >> 4;
    int nn = i & 15;
    ldsWl[i] = Wl[kk * D + n_base + nn];
    ldsWr[i] = Wr[kk * D + n_base + nn];
  }
  __syncthreads();

  const int wave = threadIdx.x >> 5;
  const int lane = threadIdx.x & 31;
  const int half = lane >> 4;      // 0: K=0,1  1: K=2,3 (A/B fragment halves)
  const int l16  = lane & 15;
  const int m_tile = blockIdx.x * 4 + wave;
  if (m_tile >= M_TILES) return;   // wave-uniform: EXEC stays all-ones for WMMA

  const int arow = m_tile * 16 + l16;              // A-fragment row for this lane
  const float rdeg = 1.0f / fmaxf(deg[arow], 1.0f);

  const float* aggRow  = sum + (size_t)arow * D;
  const float* selfRow = xin + (size_t)arow * D;

  // C/D tile: VGPR r holds (M = r + 8*half, N = l16) -> bias depends on N only
  const float bias = bl[n_base + l16];
  v8f acc;
#pragma unroll
  for (int r = 0; r < 8; ++r) acc[r] = bias;

  const int koff = 2 * half;       // this lane's K offset within the 4-wide slab
#pragma unroll 4
  for (int k = 0; k < D; k += 4) {
    // aggregate (mean) path: A = sum-row * 1/deg, B = Wl tile
    v2f a0 = *(const v2f*)(aggRow + k + koff);
    a0 *= rdeg;
    v2f b0;
    b0.x = ldsWl[(k + koff) * 16 + l16];
    b0.y = ldsWl[(k + koff + 1) * 16 + l16];
    acc = __builtin_amdgcn_wmma_f32_16x16x4_f32(
        false, a0, false, b0, (short)0, acc, false, false);

    // self path: A = x-row, B = Wr tile
    v2f a1 = *(const v2f*)(selfRow + k + koff);
    v2f b1;
    b1.x = ldsWr[(k + koff) * 16 + l16];
    b1.y = ldsWr[(k + koff + 1) * 16 + l16];
    acc = __builtin_amdgcn_wmma_f32_16x16x4_f32(
        false, a1, false, b1, (short)0, acc, false, false);
  }

  // Write back with ReLU. VGPR r -> row (m_tile*16 + r + 8*half), col n_base+l16.
#pragma unroll
  for (int r = 0; r < 8; ++r) {
    int orow = m_tile * 16 + r + 8 * half;
    out[(size_t)orow * D + n_base + l16] = fmaxf(acc[r], 0.0f);
  }
}

// ---------------------------------------------------------------------------
// kernel_launch
// inputs: x, edge_index, edge_weight(unused), W0l, b0l, W0r, W2l, b2l, W2r
// d_out:  [fea (12.8M f32)] [h (12.8M f32)]
// d_ws:   [sum 51.2MB] [deg 0.4MB]
// ---------------------------------------------------------------------------
extern "C" void kernel_launch(void* const* d_in, const int* in_sizes, int n_in,
                              void* d_out, int out_size, void* d_ws, size_t ws_size,
                              hipStream_t stream) {
  const float* x   = (const float*)d_in[0];
  const int*   ei  = (const int*)d_in[1];
  const float* W0l = (const float*)d_in[3];
  const float* b0l = (const float*)d_in[4];
  const float* W0r = (const float*)d_in[5];
  const float* W2l = (const float*)d_in[6];
  const float* b2l = (const float*)d_in[7];
  const float* W2r = (const float*)d_in[8];

  float* fea = (float*)d_out;              // first output
  float* h   = fea + FEAT_ELEMS;           // second output (mid features)

  float* sum = (float*)d_ws;               // [N,128] aggregate scratch
  float* deg = sum + FEAT_ELEMS;           // [N] degree

  const int* src = ei;                     // edge_index[0]
  const int* dst = ei + N_EDGES;           // edge_index[1]

  const int scatter_blocks = (N_EDGES * 32 + 255) / 256;   // 200000
  dim3 gemm_grid((M_TILES + 3) / 4, D / 16);               // (1563, 8)

  // ---- layer 0 ----
  zero_kernel<<<4096, 256, 0, stream>>>(sum, FEAT_ELEMS);
  zero_kernel<<<256, 256, 0, stream>>>(deg, N_NODES);
  degree_kernel<<<(N_EDGES + 255) / 256, 256, 0, stream>>>(dst, deg);
  scatter_kernel<<<scatter_blocks, 256, 0, stream>>>(x, src, dst, sum);
  sage_gemm_kernel<<<gemm_grid, 128, 0, stream>>>(sum, deg, x, W0l, b0l, W0r, h);

  // ---- layer 1 (degree unchanged, reuse) ----
  zero_kernel<<<4096, 256, 0, stream>>>(sum, FEAT_ELEMS);
  scatter_kernel<<<scatter_blocks, 256, 0, stream>>>(h, src, dst, sum);
  sage_gemm_kernel<<<gemm_grid, 128, 0, stream>>>(sum, deg, h, W2l, b2l, W2r, fea);
}